// RobustLoss_37452114821199
// MI455X (gfx1250) — compile-verified
//
#include <hip/hip_runtime.h>
#include <hip/hip_bf16.h>
#include <stdint.h>

#define HH 4096
#define WW 4096
#define N_PIX (HH * WW)            // 16777216
#define RANK_LO 8388607u           // N/2 - 1
#define RANK_HI 8388608u           // N/2

typedef __attribute__((ext_vector_type(2))) float v2f;
typedef __attribute__((ext_vector_type(8))) float v8f;

#define USE_ASYNC_LDS 1

// ---------- monotonic float <-> uint mapping for radix select ----------
__device__ __forceinline__ unsigned mono(float f) {
    unsigned b = __float_as_uint(f);
    return (b & 0x80000000u) ? ~b : (b | 0x80000000u);
}
__device__ __forceinline__ float mono_inv(unsigned u) {
    unsigned b = (u & 0x80000000u) ? (u & 0x7fffffffu) : ~u;
    return __uint_as_float(b);
}

// ---------- Stage 0a: 65536-bin histogram of high 16 bits --------------
// CDNA5: 320KB LDS per WGP -> a full 256KB histogram fits in LDS.
__global__ void hist_hi_kernel(const float* __restrict__ r0,
                               unsigned* __restrict__ ghist) {
    extern __shared__ unsigned lh[];           // 65536 bins (256 KB dynamic LDS)
    for (int b = threadIdx.x; b < 65536; b += blockDim.x) lh[b] = 0u;
    __syncthreads();
    int stride = gridDim.x * blockDim.x;
    for (int i = blockIdx.x * blockDim.x + threadIdx.x; i < N_PIX; i += stride) {
        unsigned u = mono(r0[i]);
        atomicAdd(&lh[u >> 16], 1u);           // ds_add
    }
    __syncthreads();
    for (int b = threadIdx.x; b < 65536; b += blockDim.x) {
        unsigned v = lh[b];
        if (v) atomicAdd(&ghist[b], v);        // sparse flush
    }
}

// ---------- Stage 0b: find the hi-16 bin holding each target rank ------
__global__ void select_hi_kernel(const unsigned* __restrict__ hist,
                                 unsigned* __restrict__ sel) {
    __shared__ unsigned part[256];
    __shared__ unsigned excl[256];
    int t = threadIdx.x;
    unsigned s = 0;
    for (int b = 0; b < 256; ++b) s += hist[t * 256 + b];
    part[t] = s;
    __syncthreads();
    if (t == 0) {
        unsigned c = 0;
        for (int i = 0; i < 256; ++i) { excl[i] = c; c += part[i]; }
    }
    __syncthreads();
    unsigned ranks[2] = {RANK_LO, RANK_HI};
    unsigned cum = excl[t];
    for (int b = 0; b < 256; ++b) {
        unsigned h = hist[t * 256 + b];
        unsigned bin = (unsigned)(t * 256 + b);
        for (int rr = 0; rr < 2; ++rr) {
            if (h && ranks[rr] >= cum && ranks[rr] < cum + h) {
                sel[rr * 2 + 0] = bin;  // hi bin
                sel[rr * 2 + 1] = cum;  // count strictly below bin
            }
        }
        cum += h;
    }
}

// ---------- Stage 0c: low-16-bit histograms restricted to the bins -----
__global__ void hist_lo_kernel(const float* __restrict__ r0,
                               const unsigned* __restrict__ sel,
                               unsigned* __restrict__ lo0,
                               unsigned* __restrict__ lo1) {
    unsigned hb0 = sel[0], hb1 = sel[2];
    int stride = gridDim.x * blockDim.x;
    for (int i = blockIdx.x * blockDim.x + threadIdx.x; i < N_PIX; i += stride) {
        unsigned u = mono(r0[i]);
        unsigned hi = u >> 16;
        if (hi == hb0) atomicAdd(&lo0[u & 0xffffu], 1u);
        if (hi == hb1) atomicAdd(&lo1[u & 0xffffu], 1u);
    }
}

// ---------- Stage 0d: finish selection, median = avg of the two --------
__global__ void select_lo_kernel(const unsigned* __restrict__ lo0,
                                 const unsigned* __restrict__ lo1,
                                 const unsigned* __restrict__ sel,
                                 float* __restrict__ med) {
    __shared__ unsigned part[256];
    __shared__ unsigned excl[256];
    __shared__ float vals[2];
    int t = threadIdx.x;
    for (int rr = 0; rr < 2; ++rr) {
        const unsigned* h = rr ? lo1 : lo0;
        unsigned target = (rr ? RANK_HI : RANK_LO) - sel[rr * 2 + 1];
        unsigned s = 0;
        for (int b = 0; b < 256; ++b) s += h[t * 256 + b];
        part[t] = s;
        __syncthreads();
        if (t == 0) {
            unsigned c = 0;
            for (int i = 0; i < 256; ++i) { excl[i] = c; c += part[i]; }
        }
        __syncthreads();
        unsigned cum = excl[t];
        for (int b = 0; b < 256; ++b) {
            unsigned hb = h[t * 256 + b];
            if (hb && target >= cum && target < cum + hb) {
                unsigned u = (sel[rr * 2] << 16) | (unsigned)(t * 256 + b);
                vals[rr] = mono_inv(u);
            }
            cum += hb;
        }
        __syncthreads();
    }
    if (t == 0) med[0] = 0.5f * (vals[0] + vals[1]);
}

// ---------- Stage 1: 3x3 zero-padded box filter, async-to-LDS tiles ----
#define TY 32
#define TX 64
#define LDSW 68  // 66 used + pad
__global__ void box3_kernel(const float* __restrict__ r0,
                            const float* __restrict__ medp,
                            float* __restrict__ hn) {
    __shared__ float lds[34 * LDSW];   // sole shared object -> LDS offset 0
    int bx = blockIdx.x, by = blockIdx.y;
    int gy0 = by * TY, gx0 = bx * TX;
    int tid = threadIdx.x;
    // stage (TY+2) x (TX+2) halo tile; OOB cells = 0 (conv 'SAME' zero-pad of x)
    for (int idx = tid; idx < 34 * 66; idx += 256) {
        int r = idx / 66, c = idx % 66;
        int gy = gy0 - 1 + r;
        int gx = gx0 - 1 + c;
        if (gy >= 0 && gy < HH && gx >= 0 && gx < WW) {
#if USE_ASYNC_LDS
            unsigned ldsoff = (unsigned)((r * LDSW + c) * 4);
            unsigned long long ga =
                (unsigned long long)(uintptr_t)(r0 + (size_t)gy * WW + gx);
            asm volatile("global_load_async_to_lds_b32 %0, %1, off"
                         :: "v"(ldsoff), "v"(ga) : "memory");
#else
            lds[r * LDSW + c] = r0[(size_t)gy * WW + gx];
#endif
        } else {
            lds[r * LDSW + c] = 0.0f;
        }
    }
#if USE_ASYNC_LDS
    asm volatile("s_wait_asynccnt 0x0" ::: "memory");
#endif
    __syncthreads();

    float m = *medp;
    int tx = tid & 63;
    int tyq = tid >> 6;
    for (int k = 0; k < 8; ++k) {
        int ty = tyq * 8 + k;
        int gy = gy0 + ty, gx = gx0 + tx;
        float sum = 0.0f;
        #pragma unroll
        for (int dy = 0; dy < 3; ++dy)
            #pragma unroll
            for (int dx = 0; dx < 3; ++dx)
                sum += lds[(ty + dy) * LDSW + (tx + dx)];
        int rowtaps = 3 - (gy == 0) - (gy == HH - 1);
        int coltaps = 3 - (gx == 0) - (gx == WW - 1);
        hn[(size_t)gy * WW + gx] = (sum - m * (float)(rowtaps * coltaps)) * (1.0f / 9.0f);
    }
}

// ---------- Stage 2a: 4-row partial sums of hn -------------------------
__global__ void qsum_kernel(const float* __restrict__ hn, float* __restrict__ qsum) {
    int stride = gridDim.x * blockDim.x;
    for (int idx = blockIdx.x * blockDim.x + threadIdx.x; idx < 1024 * WW; idx += stride) {
        int q = idx >> 12, x = idx & 4095;
        size_t b = (size_t)(q * 4) * WW + x;
        qsum[idx] = hn[b] + hn[b + WW] + hn[b + 2 * WW] + hn[b + 3 * WW];
    }
}

// ---------- Stage 2b: 16-row (stride 8, edge-replicated) row pooling ---
__global__ void rowsum_kernel(const float* __restrict__ qsum,
                              const float* __restrict__ hn,
                              float* __restrict__ rowsum) {
    int stride = gridDim.x * blockDim.x;
    for (int idx = blockIdx.x * blockDim.x + threadIdx.x; idx < 512 * WW; idx += stride) {
        int i = idx >> 12, x = idx & 4095;
        float s;
        if (i == 0) {
            s = qsum[0 * WW + x] + qsum[1 * WW + x] + qsum[2 * WW + x]
              + 4.0f * hn[(size_t)0 * WW + x];
        } else if (i == 511) {
            s = qsum[1021 * WW + x] + qsum[1022 * WW + x] + qsum[1023 * WW + x]
              + 4.0f * hn[(size_t)4095 * WW + x];
        } else {
            s = qsum[(2 * i - 1) * WW + x] + qsum[(2 * i) * WW + x]
              + qsum[(2 * i + 1) * WW + x] + qsum[(2 * i + 2) * WW + x];
        }
        rowsum[idx] = s;
    }
}

// ---------- Stage 2c: column pooling as GEMM via V_WMMA_F32_16X16X4_F32 -
// pooled[i,j] = (1/256) * sum_c w(c,j) * rowsum[i,c]; w includes edge-clamp
// multiplicities of the 16-wide stride-8 window on the edge-padded image.
// Branchless: pure selects -> v_cndmask, no exec-mask control flow.
__device__ __forceinline__ float pool_w(int c, int j) {
    int lo = 8 * j - 4;
    int w_in    = (c >= lo && c <= lo + 15) ? 1 : 0;
    int w_left  = min(16, max(0, 1 - lo));
    int w_right = min(16, max(0, lo + 16 - WW));
    int w = (c == 0) ? w_left : ((c == WW - 1) ? w_right : w_in);
    w = (c < 0 || c >= WW) ? 0 : w;
    return (float)w;
}

__global__ void pool_wmma_kernel(const float* __restrict__ rowsum,
                                 float* __restrict__ pooled) {
    // 32x32 tiles of 16x16; one wave (32 lanes) per tile; 8 waves/block
    int wave = threadIdx.x >> 5;
    int lane = threadIdx.x & 31;
    int tile = blockIdx.x * 8 + wave;          // 0..1023
    int ti = tile >> 5, tj = tile & 31;
    int i0 = ti * 16, j0 = tj * 16;

    int half = (lane & 16) ? 1 : 0;            // lane group
    int koff = half * 2;                       // A/B K offset within 4-step
    int arow = i0 + (lane & 15);               // A: row per lane
    int jg   = j0 + (lane & 15);               // B: output column per lane
    const float* rbase = rowsum + (size_t)arow * WW;

    v8f acc = {};
    int ksbase = 8 * j0 - 4;                   // first input column of window
    for (int s = 0; s < 34; ++s) {             // K = 136 input cols, 4 per step
        int c0 = ksbase + 4 * s + koff;        // even -> (c0,c1) pair 8B-aligned
        int c1 = c0 + 1;
        // (c0,c1) are always both-in-range or both-OOB (4s+koff is even), and
        // OOB taps carry weight 0, so clamp the address and load the pair
        // unconditionally as one b64.
        int c0c = min(max(c0, 0), WW - 2);
        v2f a = *(const v2f*)(rbase + c0c);
        v2f b;
        b.x = pool_w(c0, jg);
        b.y = pool_w(c1, jg);
        acc = __builtin_amdgcn_wmma_f32_16x16x4_f32(
            false, a, false, b, (short)0, acc, false, false);
    }
    // C/D layout: VGPR v -> row i0+v (lanes 0-15) / i0+v+8 (lanes 16-31)
    int prow = i0 + half * 8;
    #pragma unroll
    for (int v = 0; v < 8; ++v)
        pooled[(size_t)(prow + v) * 512 + j0 + (lane & 15)] = acc[v] * (1.0f / 256.0f);
}

// ---------- Stage 3: fused upsample + mask + linear + sigmoid ----------
__global__ void final_kernel(const float* __restrict__ r0,
                             const float* __restrict__ r1,
                             const float* __restrict__ hn,
                             const float* __restrict__ pooled,
                             const float* __restrict__ medp,
                             const float* __restrict__ w1,
                             const float* __restrict__ b1,
                             float* __restrict__ out) {
    float m = *medp, w00 = w1[0], w01 = w1[1], bb = b1[0];
    int stride = gridDim.x * blockDim.x;
    for (int i = blockIdx.x * blockDim.x + threadIdx.x; i < N_PIX; i += stride) {
        int y = i >> 12, x = i & 4095;
        float mask = pooled[(size_t)(y >> 3) * 512 + (x >> 3)] + hn[i] + (r0[i] - m);
        float z = r1[i] * w00 + mask * w01 + bb;
        out[i] = 1.0f / (1.0f + expf(-z));
        out[(size_t)N_PIX + i] = mask;
    }
}

extern "C" void kernel_launch(void* const* d_in, const int* in_sizes, int n_in,
                              void* d_out, int out_size, void* d_ws, size_t ws_size,
                              hipStream_t stream) {
    (void)in_sizes; (void)n_in; (void)out_size; (void)ws_size;
    const float* residuals = (const float*)d_in[0];   // (2,1,4096,4096)
    const float* w1 = (const float*)d_in[1];          // (1,2)
    const float* b1 = (const float*)d_in[2];          // (1,)
    const float* r0 = residuals;
    const float* r1 = residuals + (size_t)N_PIX;
    float* out = (float*)d_out;                       // [gated | mask]

    // workspace layout (floats / u32)
    unsigned* hist_hi = (unsigned*)d_ws;              // 65536
    unsigned* lo0 = hist_hi + 65536;                  // 65536
    unsigned* lo1 = lo0 + 65536;                      // 65536
    unsigned* sel = lo1 + 65536;                      // 8
    float* med = (float*)(sel + 8);                   // 1
    float* base = (float*)d_ws;
    float* hn     = base + 262144;                    // 4096*4096
    float* qsum   = hn + (size_t)N_PIX;               // 1024*4096
    float* rowsum = qsum + (size_t)1024 * WW;         // 512*4096
    float* pooled = rowsum + (size_t)512 * WW;        // 512*512

    hipMemsetAsync(d_ws, 0, 262144 * sizeof(float), stream);  // hists + scalars

    hist_hi_kernel<<<1024, 256, 65536 * sizeof(unsigned), stream>>>(r0, hist_hi);
    select_hi_kernel<<<1, 256, 0, stream>>>(hist_hi, sel);
    hist_lo_kernel<<<2048, 256, 0, stream>>>(r0, sel, lo0, lo1);
    select_lo_kernel<<<1, 256, 0, stream>>>(lo0, lo1, sel, med);
    box3_kernel<<<dim3(WW / TX, HH / TY), 256, 0, stream>>>(r0, med, hn);
    qsum_kernel<<<4096, 256, 0, stream>>>(hn, qsum);
    rowsum_kernel<<<2048, 256, 0, stream>>>(qsum, hn, rowsum);
    pool_wmma_kernel<<<128, 256, 0, stream>>>(rowsum, pooled);
    final_kernel<<<4096, 256, 0, stream>>>(r0, r1, hn, pooled, med, w1, b1, out);
}